// PhysicsConstraintLoss_58909771432750
// MI455X (gfx1250) — compile-verified
//
#include <hip/hip_runtime.h>
#include <hip/hip_bf16.h>

typedef __attribute__((ext_vector_type(2))) float v2f;
typedef __attribute__((ext_vector_type(4))) float v4f;
typedef __attribute__((ext_vector_type(8))) float v8f;
typedef __attribute__((ext_vector_type(4))) int   v4i;

#define BATCH   8
#define NNODES  100000
#define NEDGES  3200000
#define NRES    64

// d_ws layout (floats): [0]=continuity sq-sum, [1]=boundary sq-sum, [2..63] pad,
//                       [64 .. 64+BATCH*NNODES) = net_flows accumulator
#define WS_NET_OFF 64

// -------- wave32 sum-reduction via V_WMMA_F32_16X16X4_F32 (B = ones) --------
// A (16x4 f32): lane l supplies VGPR0 = acc, VGPR1 = 0  ->  rowsum[m] = acc(m) + acc(m+16)
// D (16x16 f32): lane 0 VGPRs 0..7 = rows 0..7 (col 0); lane 16 VGPRs 0..7 = rows 8..15.
// Summing the 8 D elements on lanes 0 and 16 and adding both gives the full wave sum.
// (clang's atomic optimizer further folds the two lane-atomics into one per wave.)
__device__ __forceinline__ void wave_sum_wmma_atomic(float acc, float* target, int lane) {
    v2f a;    a.x = acc; a.y = 0.0f;
    v2f ones; ones.x = 1.0f; ones.y = 1.0f;
    v8f cz = {};
    v8f d = __builtin_amdgcn_wmma_f32_16x16x4_f32(
        /*neg_a=*/false, a, /*neg_b=*/false, ones,
        /*c_mod=*/(short)0, cz, /*reuse_a=*/false, /*reuse_b=*/false);
    float s = ((d[0] + d[1]) + (d[2] + d[3])) + ((d[4] + d[5]) + (d[6] + d[7]));
    if (lane == 0 || lane == 16) {
        unsafeAtomicAdd(target, s);   // global_atomic_add_f32
    }
}

// -------- kernel 0: zero workspace (accumulators + net_flows) --------
__global__ void k_zero(float* __restrict__ ws, int n) {
    int i = blockIdx.x * blockDim.x + threadIdx.x;
    int stride = gridDim.x * blockDim.x;
    for (; i < n; i += stride) ws[i] = 0.0f;
}

// -------- kernel 1: flows + atomic scatter of net flows (4 edges / thread) --------
// Streaming data (indices, attrs in; flows out) uses non-temporal hints so the
// 192MB L2 stays dedicated to the hot gather/scatter arrays (heads/net, 6.4MB).
__global__ void k_edges(const float* __restrict__ heads,
                        const int*   __restrict__ src,
                        const int*   __restrict__ dst,
                        const float* __restrict__ eattr,
                        float*       __restrict__ flows,   // = d_out + 3
                        float*       __restrict__ net)
{
    int g  = blockIdx.x * blockDim.x + threadIdx.x;   // group of 4 consecutive edges
    int e0 = g * 4;
    if (e0 >= NEDGES) return;                          // E % 4 == 0 -> no ragged tail

    v4i s4  = __builtin_nontemporal_load((const v4i*)(src + e0));             // b128 NT
    v4i d4  = __builtin_nontemporal_load((const v4i*)(dst + e0));
    v4f ea0 = __builtin_nontemporal_load((const v4f*)(eattr + 2 * (size_t)e0));     // (c0,r0,c1,r1)
    v4f ea1 = __builtin_nontemporal_load((const v4f*)(eattr + 2 * (size_t)e0 + 4)); // (c2,r2,c3,r3)

    const int   s[4] = { s4.x, s4.y, s4.z, s4.w };
    const int   d[4] = { d4.x, d4.y, d4.z, d4.w };
    const float c[4] = { ea0.x, ea0.z, ea1.x, ea1.z };

    #pragma unroll
    for (int b = 0; b < BATCH; ++b) {
        const float* hb = heads + (size_t)b * NNODES;   // 3.2 MB, L2-resident gathers
        float*       nb = net   + (size_t)b * NNODES;   // 3.2 MB, L2 fp32 atomics
        float*       fb = flows + (size_t)b * NEDGES + e0;
        #pragma unroll
        for (int j = 0; j < 4; ++j) {
            float f = c[j] * (hb[s[j]] - hb[d[j]]);
            __builtin_nontemporal_store(f, fb + j);     // coalesced NT b32 store stream
            unsafeAtomicAdd(nb + s[j],  f);
            unsafeAtomicAdd(nb + d[j], -f);
        }
    }
}

// -------- kernel 2: continuity loss sum((net - demand)^2), WMMA wave reduce --------
__global__ void k_continuity(const float* __restrict__ net,
                             const float* __restrict__ demands,
                             float*       __restrict__ accs)
{
    const v4f* n4 = (const v4f*)net;
    const v4f* d4 = (const v4f*)demands;
    const int NV = (BATCH * NNODES) / 4;   // 200000, exact
    int tid    = blockIdx.x * blockDim.x + threadIdx.x;
    int stride = gridDim.x * blockDim.x;
    float acc = 0.0f;
    for (int i = tid; i < NV; i += stride) {
        v4f n = __builtin_nontemporal_load(n4 + i);   // b128, single-use data
        v4f m = __builtin_nontemporal_load(d4 + i);
        v4f v = n - m;
        acc = fmaf(v.x, v.x, acc);
        acc = fmaf(v.y, v.y, acc);
        acc = fmaf(v.z, v.z, acc);
        acc = fmaf(v.w, v.w, acc);
    }
    // loop has reconverged: EXEC is all-1s here (required by WMMA)
    wave_sum_wmma_atomic(acc, accs + 0, threadIdx.x & 31);
}

// -------- kernel 3: boundary loss over 8*64 reservoir samples --------
__global__ void k_boundary(const float* __restrict__ heads,
                           const int*   __restrict__ res_nodes,
                           const float* __restrict__ res_head,
                           float*       __restrict__ accs)
{
    int t = threadIdx.x;          // single block of 512 = BATCH * NRES threads
    int b = t >> 6;
    int k = t & 63;
    float v = heads[(size_t)b * NNODES + res_nodes[k]] - res_head[0];
    wave_sum_wmma_atomic(v * v, accs + 1, t & 31);
}

// -------- kernel 4: finalize the three scalars --------
__global__ void k_final(const float* __restrict__ accs, float* __restrict__ out) {
    if (threadIdx.x == 0 && blockIdx.x == 0) {
        float cont = accs[0] * (1.0f / (float)(BATCH * NNODES));
        float bnd  = accs[1] * (1.0f / (float)(BATCH * NRES));
        out[0] = cont;
        out[1] = bnd;
        out[2] = cont + bnd;      // LAMBDA_PHYSICS = 1.0
    }
}

extern "C" void kernel_launch(void* const* d_in, const int* in_sizes, int n_in,
                              void* d_out, int out_size, void* d_ws, size_t ws_size,
                              hipStream_t stream) {
    const float* node_heads = (const float*)d_in[0];          // (B, N)
    const float* demands    = (const float*)d_in[1];          // (B, N)
    const int*   edge_index = (const int*)  d_in[2];          // (2, E)
    const float* edge_attr  = (const float*)d_in[3];          // (E, 2)
    const int*   res_nodes  = (const int*)  d_in[4];          // (64,)
    const float* res_head   = (const float*)d_in[5];          // (1,)

    const int* src = edge_index;
    const int* dst = edge_index + NEDGES;

    float* out   = (float*)d_out;
    float* flows = out + 3;                   // outputs: [cont, bnd, total, flows(B*E)]
    float* wsf   = (float*)d_ws;
    float* accs  = wsf;
    float* net   = wsf + WS_NET_OFF;

    // 1) zero accumulators + net_flows (re-done every call: harness doesn't re-poison)
    int nzero = WS_NET_OFF + BATCH * NNODES;  // 800064 floats
    k_zero<<<1024, 256, 0, stream>>>(wsf, nzero);

    // 2) flows + scatter (E/4 = 800000 threads exactly)
    k_edges<<<(NEDGES / 4 + 255) / 256, 256, 0, stream>>>(node_heads, src, dst,
                                                          edge_attr, flows, net);

    // 3) continuity reduction (fixed grid; every wave fully active at the WMMA)
    k_continuity<<<1024, 256, 0, stream>>>(net, demands, accs);

    // 4) boundary reduction (one block, 512 threads = 16 full waves)
    k_boundary<<<1, BATCH * NRES, 0, stream>>>(node_heads, res_nodes, res_head, accs);

    // 5) finalize scalars
    k_final<<<1, 64, 0, stream>>>(accs, out);
}